// DecoderAttention_88742614270479
// MI455X (gfx1250) — compile-verified
//
#include <hip/hip_runtime.h>
#include <math.h>

#define MAX_SEQ   1024
#define NUM_HEADS 16
#define NUM_KV    4
#define HEAD_DIM  128
#define HIDDEN    2048
#define GROUPS    4
#define BATCH     32
#define EPS       1e-6f
#define LN_THETA  13.815510557964274f   /* ln(1e6) */
#define SCALE     0.08838834764831845f  /* 1/sqrt(128) */

typedef __attribute__((ext_vector_type(2))) float v2f;
typedef __attribute__((ext_vector_type(8))) float v8f;

// ---------------------------------------------------------------------------
// C[M x N] = X[M x K] @ W[N x K]^T  using V_WMMA_F32_16X16X4_F32.
// One wave (32 threads) per 16x16 C tile. Per-lane operand layout (ISA 7.12.2):
//   A (16x4):  lane L<16 -> row M=L,   VGPR{0,1} = K{0,1};  lane>=16 -> K{2,3}
//   B (4x16):  lane L<16 -> col N=L,   VGPR{0,1} = K{0,1};  lane>=16 -> K{2,3}
//   C/D:       lane L<16 -> col N=L, VGPR r -> row M=r; lane>=16 -> M=r+8
// Both A and B fetch a contiguous K-pair per lane -> single b64 load each.
// ---------------------------------------------------------------------------
__global__ void gemm_xwT_wmma(const float* __restrict__ X,
                              const float* __restrict__ W,
                              float* __restrict__ C,
                              int N, int K) {
  const int n0   = blockIdx.x * 16;
  const int m0   = blockIdx.y * 16;
  const int lane = threadIdx.x;        // 0..31
  const int row  = lane & 15;
  const int koff = (lane >> 4) * 2;    // 0 or 2

  const float* xrow = X + (size_t)(m0 + row) * K + koff;
  const float* wrow = W + (size_t)(n0 + row) * K + koff;

  v8f acc = {};
  for (int k0 = 0; k0 < K; k0 += 4) {
    v2f a = *(const v2f*)(xrow + k0);
    v2f b = *(const v2f*)(wrow + k0);
    acc = __builtin_amdgcn_wmma_f32_16x16x4_f32(
        /*neg_a=*/false, a, /*neg_b=*/false, b,
        /*c_mod=*/(short)0, acc, /*reuse_a=*/false, /*reuse_b=*/false);
  }

  const int col   = n0 + row;
  const int rbase = m0 + (lane >> 4) * 8;
#pragma unroll
  for (int r = 0; r < 8; ++r)
    C[(size_t)(rbase + r) * N + col] = acc[r];
}

// ---------------------------------------------------------------------------
// Fused RMSNorm + RoPE over one 128-dim head per block (64 threads).
// t layout: [nblocks][128], in-place.
// ---------------------------------------------------------------------------
__global__ void rmsnorm_rope(float* __restrict__ t,
                             const float* __restrict__ w,
                             const float* __restrict__ position) {
  float* h = t + (size_t)blockIdx.x * HEAD_DIM;
  const int tid = threadIdx.x;  // 0..63

  float x1 = h[tid];
  float x2 = h[tid + 64];

  __shared__ float red[64];
  red[tid] = x1 * x1 + x2 * x2;
  __syncthreads();
#pragma unroll
  for (int off = 32; off > 0; off >>= 1) {
    if (tid < off) red[tid] += red[tid + off];
    __syncthreads();
  }
  const float norm = rsqrtf(red[0] * (1.0f / HEAD_DIM) + EPS);

  const float pos  = position[0];
  const float invf = expf(-(float)tid * (LN_THETA / 64.0f));
  const float fr   = pos * invf;
  const float c = cosf(fr), s = sinf(fr);

  const float y1 = x1 * norm * w[tid];
  const float y2 = x2 * norm * w[tid + 64];
  h[tid]      = y1 * c - y2 * s;
  h[tid + 64] = y2 * c + y1 * s;
}

// ---------------------------------------------------------------------------
// Attention for one (batch, kv-head): 4 grouped q-heads, full softmax over
// MAX_SEQ with masked positions skipped (exact: exp underflows to 0 in fp32).
// Row s == pos is sourced from the freshly projected k/v (onehot splice).
// 256 threads.
// ---------------------------------------------------------------------------
__global__ void attn_decode(const float* __restrict__ q,
                            const float* __restrict__ knew,
                            const float* __restrict__ vnew,
                            const float* __restrict__ k_cache,
                            const float* __restrict__ v_cache,
                            const float* __restrict__ mask,
                            const float* __restrict__ position,
                            float* __restrict__ attn_out) {
  const int kh  = blockIdx.x;          // 0..3
  const int b   = blockIdx.y;          // 0..31
  const int tid = threadIdx.x;         // 0..255
  const int pos = (int)position[0];

  __shared__ float sc[GROUPS][MAX_SEQ];   // scores -> probabilities (16 KB)
  __shared__ float qs[GROUPS][HEAD_DIM];  // 2 KB
  __shared__ float red[256];
  __shared__ float hmax[GROUPS], hsum[GROUPS];
  __shared__ float part[GROUPS][HEAD_DIM];

  const float* qh   = q + (size_t)(b * NUM_HEADS + kh * GROUPS) * HEAD_DIM;
  const float* kb   = k_cache + ((size_t)(b * NUM_KV + kh)) * MAX_SEQ * HEAD_DIM;
  const float* vb   = v_cache + ((size_t)(b * NUM_KV + kh)) * MAX_SEQ * HEAD_DIM;
  const float* krow_new = knew + (size_t)(b * NUM_KV + kh) * HEAD_DIM;
  const float* vrow_new = vnew + (size_t)(b * NUM_KV + kh) * HEAD_DIM;

  for (int i = tid; i < GROUPS * HEAD_DIM; i += 256)
    qs[i >> 7][i & 127] = qh[i];
  __syncthreads();

  // ---- scores: each thread owns 4 strided positions -----------------------
  for (int s = tid; s < MAX_SEQ; s += 256) {
    const float m = mask[s];
    if (m < -1e8f) {
      sc[0][s] = sc[1][s] = sc[2][s] = sc[3][s] = -3.0e38f;
    } else {
      const float* kr = (s == pos) ? krow_new : (kb + (size_t)s * HEAD_DIM);
      float d0 = 0.f, d1 = 0.f, d2 = 0.f, d3 = 0.f;
      for (int d = 0; d < HEAD_DIM; d += 4) {
        const float4 kv = *(const float4*)(kr + d);
        d0 += qs[0][d] * kv.x + qs[0][d+1] * kv.y + qs[0][d+2] * kv.z + qs[0][d+3] * kv.w;
        d1 += qs[1][d] * kv.x + qs[1][d+1] * kv.y + qs[1][d+2] * kv.z + qs[1][d+3] * kv.w;
        d2 += qs[2][d] * kv.x + qs[2][d+1] * kv.y + qs[2][d+2] * kv.z + qs[2][d+3] * kv.w;
        d3 += qs[3][d] * kv.x + qs[3][d+1] * kv.y + qs[3][d+2] * kv.z + qs[3][d+3] * kv.w;
      }
      sc[0][s] = d0 * SCALE + m;
      sc[1][s] = d1 * SCALE + m;
      sc[2][s] = d2 * SCALE + m;
      sc[3][s] = d3 * SCALE + m;
    }
  }
  __syncthreads();

  // ---- softmax: max, exp, sum (block reductions) --------------------------
#pragma unroll
  for (int h = 0; h < GROUPS; ++h) {
    float lm = -3.0e38f;
    for (int s = tid; s < MAX_SEQ; s += 256) lm = fmaxf(lm, sc[h][s]);
    red[tid] = lm;
    __syncthreads();
    for (int off = 128; off > 0; off >>= 1) {
      if (tid < off) red[tid] = fmaxf(red[tid], red[tid + off]);
      __syncthreads();
    }
    if (tid == 0) hmax[h] = red[0];
    __syncthreads();
  }
#pragma unroll
  for (int h = 0; h < GROUPS; ++h) {
    const float m = hmax[h];
    float ls = 0.f;
    for (int s = tid; s < MAX_SEQ; s += 256) {
      const float p = expf(sc[h][s] - m);   // masked -> exp(-huge) == 0.0f
      sc[h][s] = p;
      ls += p;
    }
    red[tid] = ls;
    __syncthreads();
    for (int off = 128; off > 0; off >>= 1) {
      if (tid < off) red[tid] += red[tid + off];
      __syncthreads();
    }
    if (tid == 0) hsum[h] = red[0];
    __syncthreads();
  }

  // ---- out[h][d] = sum_s p[h][s] * v[s][d]; 2-way split over s ------------
  const int d  = tid & 127;
  const int sh = tid >> 7;     // 0 or 1
  float acc0 = 0.f, acc1 = 0.f, acc2 = 0.f, acc3 = 0.f;
  for (int s = sh * (MAX_SEQ / 2); s < (sh + 1) * (MAX_SEQ / 2); ++s) {
    if (mask[s] < -1e8f) continue;  // exact-zero weight, skip the V read
    const float* vr = (s == pos) ? vrow_new : (vb + (size_t)s * HEAD_DIM);
    const float vv = vr[d];
    acc0 += sc[0][s] * vv;
    acc1 += sc[1][s] * vv;
    acc2 += sc[2][s] * vv;
    acc3 += sc[3][s] * vv;
  }
  if (sh == 1) {
    part[0][d] = acc0; part[1][d] = acc1; part[2][d] = acc2; part[3][d] = acc3;
  }
  __syncthreads();
  if (sh == 0) {
    float* ob = attn_out + (size_t)b * (NUM_HEADS * HEAD_DIM)
                         + (size_t)(kh * GROUPS) * HEAD_DIM;
    ob[0 * HEAD_DIM + d] = (acc0 + part[0][d]) / hsum[0];
    ob[1 * HEAD_DIM + d] = (acc1 + part[1][d]) / hsum[1];
    ob[2 * HEAD_DIM + d] = (acc2 + part[2][d]) / hsum[2];
    ob[3 * HEAD_DIM + d] = (acc3 + part[3][d]) / hsum[3];
  }
}

// ---------------------------------------------------------------------------
extern "C" void kernel_launch(void* const* d_in, const int* in_sizes, int n_in,
                              void* d_out, int out_size, void* d_ws, size_t ws_size,
                              hipStream_t stream) {
  const float* x        = (const float*)d_in[0];   // (32,1,2048)
  const float* position = (const float*)d_in[1];   // (1,)
  const float* mask     = (const float*)d_in[2];   // (1,1,1,1024)
  const float* k_cache  = (const float*)d_in[3];   // (32,4,1024,128)
  const float* v_cache  = (const float*)d_in[4];   // (32,4,1024,128)
  /* d_in[5] = onehot: handled implicitly via s==pos substitution */
  const float* wq       = (const float*)d_in[6];   // (2048,2048)
  const float* wk       = (const float*)d_in[7];   // (512,2048)
  const float* wv       = (const float*)d_in[8];   // (512,2048)
  const float* wo       = (const float*)d_in[9];   // (2048,2048)
  const float* q_norm_w = (const float*)d_in[10];  // (128,)
  const float* k_norm_w = (const float*)d_in[11];  // (128,)
  float* out = (float*)d_out;                      // (32,1,2048)

  char* ws = (char*)d_ws;
  float* q_proj   = (float*)(ws + 0);                         // 32*2048 f32
  float* k_proj   = (float*)(ws + (size_t)BATCH*HIDDEN*4);    // 32*512
  float* v_proj   = (float*)(ws + (size_t)BATCH*HIDDEN*4 + (size_t)BATCH*NUM_KV*HEAD_DIM*4);
  float* attn_out = (float*)(ws + (size_t)BATCH*HIDDEN*4 + 2*(size_t)BATCH*NUM_KV*HEAD_DIM*4);

  // 1-3: QKV projections (WMMA f32 16x16x4)
  gemm_xwT_wmma<<<dim3(HIDDEN/16, BATCH/16), 32, 0, stream>>>(x, wq, q_proj, HIDDEN, HIDDEN);
  gemm_xwT_wmma<<<dim3((NUM_KV*HEAD_DIM)/16, BATCH/16), 32, 0, stream>>>(x, wk, k_proj, NUM_KV*HEAD_DIM, HIDDEN);
  gemm_xwT_wmma<<<dim3((NUM_KV*HEAD_DIM)/16, BATCH/16), 32, 0, stream>>>(x, wv, v_proj, NUM_KV*HEAD_DIM, HIDDEN);

  // 4-5: per-head RMSNorm + RoPE (in place)
  rmsnorm_rope<<<BATCH * NUM_HEADS, 64, 0, stream>>>(q_proj, q_norm_w, position);
  rmsnorm_rope<<<BATCH * NUM_KV,    64, 0, stream>>>(k_proj, k_norm_w, position);

  // 6: fused decode attention
  attn_decode<<<dim3(NUM_KV, BATCH), 256, 0, stream>>>(
      q_proj, k_proj, v_proj, k_cache, v_cache, mask, position, attn_out);

  // 7: output projection -> d_out
  gemm_xwT_wmma<<<dim3(HIDDEN/16, BATCH/16), 32, 0, stream>>>(attn_out, wo, out, HIDDEN, HIDDEN);
}